// Model_50096498541117
// MI455X (gfx1250) — compile-verified
//
#include <hip/hip_runtime.h>
#include <hip/hip_bf16.h>

typedef __attribute__((ext_vector_type(2))) float v2f;
typedef __attribute__((ext_vector_type(8))) float v8f;

constexpr int N_NODES = 200000;
constexpr int N_USERS = 100000;
constexpr int D       = 64;
constexpr int L       = 3;
constexpr int CAT     = 4 * D;   // 256 = D*(L+1), row stride of source_embs

// ---- init: copy emb into block 0 of source_embs + target working buffer, zero degrees ----
__global__ void init_embs(const float* __restrict__ emb, float* __restrict__ src_embs,
                          float* __restrict__ t_cur, float* __restrict__ deg_s,
                          float* __restrict__ deg_t) {
  long long idx = (long long)blockIdx.x * blockDim.x + threadIdx.x;
  long long total = (long long)N_NODES * D;
  if (idx < total) {
    int n = (int)(idx / D), f = (int)(idx % D);
    float v = emb[idx];
    src_embs[(long long)n * CAT + f] = v;
    t_cur[idx] = v;
  }
  if (idx < N_NODES) { deg_s[idx] = 0.0f; deg_t[idx] = 0.0f; }
}

// ---- degree counting (constant across layers) ----
__global__ void degrees(const int* __restrict__ es, const int* __restrict__ et,
                        float* __restrict__ deg_s, float* __restrict__ deg_t, int E) {
  int e = blockIdx.x * blockDim.x + threadIdx.x;
  if (e < E) {
    atomicAdd(&deg_s[es[E + e]], 1.0f);
    atomicAdd(&deg_t[et[E + e]], 1.0f);
  }
}

// ---- zero the scatter destinations: strided source block + dense target buffer ----
__global__ void zero_bufs(float* __restrict__ src_blk, float* __restrict__ t_next) {
  long long idx = (long long)blockIdx.x * blockDim.x + threadIdx.x;
  long long total = (long long)N_NODES * D;
  if (idx < total) {
    int n = (int)(idx / D), f = (int)(idx % D);
    src_blk[(long long)n * CAT + f] = 0.0f;
    t_next[idx] = 0.0f;
  }
}

// ---- edge scatter-add: one thread per (edge, 4 features) ----
__global__ void scatter_mean(const int* __restrict__ edges,
                             const float* __restrict__ xin, int in_stride,
                             float* __restrict__ xout, int out_stride, int E) {
  long long tid = (long long)blockIdx.x * blockDim.x + threadIdx.x;
  long long total = (long long)E * (D / 4);
  if (tid >= total) return;
  int e  = (int)(tid / (D / 4));
  int fg = (int)(tid % (D / 4)) * 4;
  int s = edges[e];
  int d = edges[E + e];
  const float4 v = *(const float4*)(xin + (long long)s * in_stride + fg);
  float* o = xout + (long long)d * out_stride + fg;
  atomicAdd(o + 0, v.x);
  atomicAdd(o + 1, v.y);
  atomicAdd(o + 2, v.z);
  atomicAdd(o + 3, v.w);
}

// ---- divide by clamped degree ----
__global__ void normalize_deg(float* __restrict__ x, int stride,
                              const float* __restrict__ deg) {
  long long idx = (long long)blockIdx.x * blockDim.x + threadIdx.x;
  long long total = (long long)N_NODES * D;
  if (idx >= total) return;
  int n = (int)(idx / D), f = (int)(idx % D);
  x[(long long)n * stride + f] *= (1.0f / fmaxf(deg[n], 1.0f));
}

// ---- user mix GEMM via V_WMMA_F32_16X16X4_F32 ----
// user_emb[u, o] = sum_k cat(u,k) * W[o,k] + b[o],  cat = [src_blk row | t_blk row], k in [0,128)
// One wave handles a 16-user tile x all 64 outputs (4 N-tiles of 16).
__global__ void mix_wmma(float* __restrict__ src_blk,   // source_embs + l*D (stride CAT)
                         float* __restrict__ t_blk,     // target buffer (stride D)
                         const float* __restrict__ W,   // [64, 128] row-major
                         const float* __restrict__ bvec // [64]
) {
  int wave = (int)((blockIdx.x * (long long)blockDim.x + threadIdx.x) >> 5);
  int lane = threadIdx.x & 31;
  constexpr int TILES = N_USERS / 16;   // 6250, exact
  if (wave >= TILES) return;            // wave-uniform: EXEC stays all-ones for WMMA
  int u0   = wave * 16;
  int half = lane >> 4;                 // 0: K offset 0,1   1: K offset 2,3
  int lid  = lane & 15;                 // A: user row in tile; B: output col in tile
  int koff = half * 2;

  const float* aSrc = src_blk + (long long)(u0 + lid) * CAT; // k in [0,64)
  const float* aTgt = t_blk   + (long long)(u0 + lid) * D;   // k-64 in [0,64)

  v8f acc[4];
#pragma unroll
  for (int nt = 0; nt < 4; ++nt) {
    float b = bvec[nt * 16 + lid];
    acc[nt] = (v8f){b, b, b, b, b, b, b, b};
  }

#pragma unroll 4
  for (int k0 = 0; k0 < 2 * D; k0 += 4) {
    int kk = k0 + koff;
    v2f a;
    if (k0 < D) { a.x = aSrc[kk];     a.y = aSrc[kk + 1];     }
    else        { a.x = aTgt[kk - D]; a.y = aTgt[kk - D + 1]; }
#pragma unroll
    for (int nt = 0; nt < 4; ++nt) {
      const float* wr = W + (long long)(nt * 16 + lid) * (2 * D);
      v2f b; b.x = wr[kk]; b.y = wr[kk + 1];
      acc[nt] = __builtin_amdgcn_wmma_f32_16x16x4_f32(
          false, a, false, b, (short)0, acc[nt], false, false);
    }
  }

  // C/D layout: VGPR r -> row r (lanes 0-15) / row r+8 (lanes 16-31), col = lane&15
#pragma unroll
  for (int r = 0; r < 8; ++r) {
    long long u = u0 + r + half * 8;
#pragma unroll
    for (int nt = 0; nt < 4; ++nt) {
      float v = acc[nt][r];
      src_blk[u * CAT + nt * 16 + lid] = v;   // replace user rows in source stream
      t_blk  [u * D   + nt * 16 + lid] = v;   // replace user rows in target stream
    }
  }
}

// ---- link prediction: one wave per link ----
__global__ void pred_links(const float* __restrict__ src_embs, const int* __restrict__ link,
                           const float* __restrict__ pw, const float* __restrict__ pb,
                           float* __restrict__ out, int NL) {
  int wave = (int)((blockIdx.x * (long long)blockDim.x + threadIdx.x) >> 5);
  int lane = threadIdx.x & 31;
  if (wave >= NL) return;
  long long u = link[wave];
  long long v = link[NL + wave];
  const float* ue = src_embs + u * CAT;
  const float* ie = src_embs + v * CAT;
  float s = 0.0f;
#pragma unroll
  for (int k = lane; k < CAT; k += 32) {
    s += ue[k] * pw[k];
    s += ie[k] * pw[CAT + k];
  }
  for (int off = 16; off > 0; off >>= 1) s += __shfl_down(s, off, 32);
  if (lane == 0) {
    float x = s + pb[0];
    x = (x >= 0.0f) ? x : 0.01f * x;          // leaky_relu
    out[wave] = 1.0f / (1.0f + __expf(-x));   // sigmoid
  }
}

extern "C" void kernel_launch(void* const* d_in, const int* in_sizes, int n_in,
                              void* d_out, int out_size, void* d_ws, size_t ws_size,
                              hipStream_t stream) {
  const int*   es     = (const int*)d_in[0];   // source_edge_index [2,E]
  const int*   et     = (const int*)d_in[1];   // target_edge_index [2,E]
  const int*   link   = (const int*)d_in[2];   // link [2,NLINK]
  const float* emb    = (const float*)d_in[3]; // [N,64]
  const float* mix_w  = (const float*)d_in[4]; // [3,64,128]
  const float* mix_b  = (const float*)d_in[5]; // [3,64]
  const float* pred_w = (const float*)d_in[6]; // [1,512]
  const float* pred_b = (const float*)d_in[7]; // [1]

  const int E  = in_sizes[0] / 2;
  const int NL = in_sizes[2] / 2;

  // workspace layout (floats): src_embs[N,256] | t_a[N,64] | t_b[N,64] | deg_s[N] | deg_t[N]
  float* ws       = (float*)d_ws;
  float* src_embs = ws;
  float* t_a      = src_embs + (size_t)N_NODES * CAT;
  float* t_b      = t_a + (size_t)N_NODES * D;
  float* deg_s    = t_b + (size_t)N_NODES * D;
  float* deg_t    = deg_s + N_NODES;

  const int blk = 256;
  const long long nd = (long long)N_NODES * D;
  const int nd_blocks = (int)((nd + blk - 1) / blk);

  init_embs<<<nd_blocks, blk, 0, stream>>>(emb, src_embs, t_a, deg_s, deg_t);
  degrees<<<(E + blk - 1) / blk, blk, 0, stream>>>(es, et, deg_s, deg_t, E);

  float* t_cur  = t_a;
  float* t_next = t_b;
  for (int l = 1; l <= L; ++l) {
    float* src_prev = src_embs + (size_t)(l - 1) * D;  // previous layer column block
    float* src_cur  = src_embs + (size_t)l * D;        // this layer column block

    zero_bufs<<<nd_blocks, blk, 0, stream>>>(src_cur, t_next);

    const long long sc = (long long)E * (D / 4);
    const int sc_blocks = (int)((sc + blk - 1) / blk);
    scatter_mean<<<sc_blocks, blk, 0, stream>>>(es, src_prev, CAT, src_cur, CAT, E);
    scatter_mean<<<sc_blocks, blk, 0, stream>>>(et, t_cur, D, t_next, D, E);

    normalize_deg<<<nd_blocks, blk, 0, stream>>>(src_cur, CAT, deg_s);
    normalize_deg<<<nd_blocks, blk, 0, stream>>>(t_next, D, deg_t);

    const long long mix_threads = (long long)(N_USERS / 16) * 32;
    mix_wmma<<<(int)((mix_threads + blk - 1) / blk), blk, 0, stream>>>(
        src_cur, t_next,
        mix_w + (size_t)(l - 1) * D * 2 * D,
        mix_b + (size_t)(l - 1) * D);

    float* tmp = t_cur; t_cur = t_next; t_next = tmp;
  }

  const long long pr_threads = (long long)NL * 32;
  pred_links<<<(int)((pr_threads + blk - 1) / blk), blk, 0, stream>>>(
      src_embs, link, pred_w, pred_b, (float*)d_out, NL);
}